// FieldAwareFactorizationMachine_22136261443936
// MI455X (gfx1250) — compile-verified
//
#include <hip/hip_runtime.h>

// FFM pairwise interactions for MI455X (gfx1250, wave32).
// out[b, p(i,j), :] = emb[j][x[b,i]+i*V][:] * emb[i][x[b,j]+j*V][:]
// Pure gather + Hadamard product: memory-bound (AI ~ 0.06 FLOP/B, ~142 MB).
// Mapping exploits: item stride 737280 ≡ 0 (mod 4) and stride/4 ≡ 0 (mod 45),
// so q and pair p are per-thread invariants; the 4 items differ only by b+=4096.
// One /45 + one triu decode per thread; 8 coalesced b128 gathers in flight;
// global_prefetch_b8 warmup; NT stores keep the reusable gather set in 192MB L2.

typedef __attribute__((ext_vector_type(4))) float fv4;

#define NUM_FIELDS 10
#define NPAIRS     45           // 10 choose 2
#define EMBED_DIM  16
#define VOCAB      100000
#define BATCH      16384

#define N_ITEMS           (BATCH * NPAIRS * 4)      // float4 chunks of output
#define ITEMS_PER_THREAD  4
#define THREADS           256
#define TOTAL_THREADS     (N_ITEMS / ITEMS_PER_THREAD)   // 737280
#define BLOCKS            (TOTAL_THREADS / THREADS)      // 2880
#define B_STRIDE          (TOTAL_THREADS / 4 / NPAIRS)   // 4096 batches per item step

static_assert(N_ITEMS % (ITEMS_PER_THREAD * THREADS) == 0, "exact grid");
static_assert(BLOCKS * THREADS * ITEMS_PER_THREAD == N_ITEMS, "full coverage");
static_assert(TOTAL_THREADS % 4 == 0, "q invariant across items");
static_assert((TOTAL_THREADS / 4) % NPAIRS == 0, "p invariant across items");
static_assert(B_STRIDE * ITEMS_PER_THREAD == BATCH, "b coverage exact");

__global__ __launch_bounds__(THREADS)
void ffm_pairs_kernel(const int* __restrict__ x,      // [BATCH, 10] field ids
                      const float* __restrict__ emb,  // [10, 10*VOCAB, 16]
                      float* __restrict__ out)        // [BATCH, 45, 16]
{
    const int tid = blockIdx.x * THREADS + threadIdx.x;

    const fv4* __restrict__ embv = reinterpret_cast<const fv4*>(emb);
    fv4*       __restrict__ outv = reinterpret_cast<fv4*>(out);

    // Per-thread invariants (identical across the 4 items).
    const int q   = tid & 3;          // which float4 of the 16-float row
    const int pb0 = tid >> 2;         // (b0, pair) flat index of item 0
    const int p   = pb0 % NPAIRS;     // pair id (invariant)
    const int b0  = pb0 / NPAIRS;     // first batch row handled by this thread

    // Branchless triu(k=1) decode: rowStart(i) = i*(19-i)/2
    const int i = (p >= 9) + (p >= 17) + (p >= 24) + (p >= 30) +
                  (p >= 35) + (p >= 39) + (p >= 42) + (p >= 44);
    const int j = i + 1 + p - ((i * (19 - i)) >> 1);

    // Row bases: term1 lives in emb field j at row x[b,i]+i*V,
    //            term2 lives in emb field i at row x[b,j]+j*V.
    const size_t baseA = (size_t)j * (size_t)(NUM_FIELDS * VOCAB) + (size_t)(i * VOCAB);
    const size_t baseB = (size_t)i * (size_t)(NUM_FIELDS * VOCAB) + (size_t)(j * VOCAB);

    // Phase 1: per-item gather addresses (index loads coalesce & hit cache).
    const fv4* ap[ITEMS_PER_THREAD];
    const fv4* bp[ITEMS_PER_THREAD];
#pragma unroll
    for (int k = 0; k < ITEMS_PER_THREAD; ++k) {
        const int xrow = (b0 + k * B_STRIDE) * NUM_FIELDS;
        const int xi = x[xrow + i];
        const int xj = x[xrow + j];
        ap[k] = embv + (baseA + (size_t)xi) * (EMBED_DIM / 4) + q;
        bp[k] = embv + (baseB + (size_t)xj) * (EMBED_DIM / 4) + q;
    }

    // Phase 2: warm the lines (gfx1250 global_prefetch_b8; no counters, no VGPRs).
#pragma unroll
    for (int k = 0; k < ITEMS_PER_THREAD; ++k) {
        __builtin_prefetch((const void*)ap[k], 0, 3);
        __builtin_prefetch((const void*)bp[k], 0, 3);
    }

    // Phase 3: all 8 b128 gathers in flight, then one wait.
    fv4 av[ITEMS_PER_THREAD], bv[ITEMS_PER_THREAD];
#pragma unroll
    for (int k = 0; k < ITEMS_PER_THREAD; ++k) {
        av[k] = ap[k][0];
        bv[k] = bp[k][0];
    }

    // Phase 4: Hadamard product; NT stores (GPU never re-reads the output) so
    // the replay-reused gather working set stays resident in the 192MB L2.
#pragma unroll
    for (int k = 0; k < ITEMS_PER_THREAD; ++k) {
        __builtin_nontemporal_store(av[k] * bv[k], &outv[tid + k * TOTAL_THREADS]);
    }
}

extern "C" void kernel_launch(void* const* d_in, const int* in_sizes, int n_in,
                              void* d_out, int out_size, void* d_ws, size_t ws_size,
                              hipStream_t stream) {
    const int*   x   = (const int*)d_in[0];    // [BATCH*10] int
    const float* emb = (const float*)d_in[1];  // [10 * 1,000,000 * 16] f32
    float*       out = (float*)d_out;          // [BATCH*45*16] f32
    (void)in_sizes; (void)n_in; (void)out_size; (void)d_ws; (void)ws_size;

    ffm_pairs_kernel<<<BLOCKS, THREADS, 0, stream>>>(x, emb, out);
}